// BoxLoss_6854767805208
// MI455X (gfx1250) — compile-verified
//
#include <hip/hip_runtime.h>

// CIoU box loss for MI455X (gfx1250, wave32).
// Memory-bound streaming kernel: 86 MB in / 8.6 MB out -> ~4 us roofline at 23.3 TB/s.
// Strategy: b128 vector loads of each box, non-temporal hints (single-use data),
// global_prefetch of the next grid-stride tile, fast v_rcp for divisions,
// deterministic two-pass reduction (block partials in d_ws, then one-block finish).

typedef float f32x4 __attribute__((ext_vector_type(4)));

#define EPSF 1e-9f
#define INV_PI2 0.4052847345693511f   // 4 / pi^2

__device__ __forceinline__ float wave_reduce_add(float v) {
    // wave32: full reduction within one wavefront
#pragma unroll
    for (int off = 16; off > 0; off >>= 1)
        v += __shfl_down(v, off, 32);
    return v;
}

__global__ __launch_bounds__(256)
void ciou_main_kernel(const f32x4* __restrict__ pb,   // predicts  [N] of float4 xyxy
                      const f32x4* __restrict__ tb,   // targets   [N] of float4 xyxy
                      const int*   __restrict__ msk,  // valid mask [N]
                      const float* __restrict__ bnorm,// box_norm  [N]
                      float*       __restrict__ iou_out, // [N] (d_out + 1)
                      float*       __restrict__ partials,// [gridDim.x] in d_ws
                      int n)
{
    const int stride = gridDim.x * blockDim.x;
    float acc = 0.0f;

    for (int i = blockIdx.x * blockDim.x + threadIdx.x; i < n; i += stride) {
        // Software prefetch of next grid-stride tile (global_prefetch_b8).
        int ip = i + stride;
        if (ip < n) {
            __builtin_prefetch(&pb[ip], 0, 0);
            __builtin_prefetch(&tb[ip], 0, 0);
        }

        // Single-use streaming data: non-temporal b128 loads.
        f32x4 a = __builtin_nontemporal_load(&pb[i]);
        f32x4 b = __builtin_nontemporal_load(&tb[i]);

        const float x1a = a.x, y1a = a.y, x2a = a.z, y2a = a.w;
        const float x1b = b.x, y1b = b.y, x2b = b.z, y2b = b.w;

        // Intersection / union
        float iw = fmaxf(fminf(x2a, x2b) - fmaxf(x1a, x1b), 0.0f);
        float ih = fmaxf(fminf(y2a, y2b) - fmaxf(y1a, y1b), 0.0f);
        float inter = iw * ih;

        float wa = x2a - x1a, ha = y2a - y1a;
        float wb = x2b - x1b, hb = y2b - y1b;
        float uni = wa * ha + wb * hb - inter;
        float iou = inter * __builtin_amdgcn_rcpf(uni + EPSF);

        // Enclosing-box diagonal^2
        float cw = fmaxf(x2a, x2b) - fminf(x1a, x1b);
        float ch = fmaxf(y2a, y2b) - fminf(y1a, y1b);
        float c2 = cw * cw + ch * ch + EPSF;

        // Center distance^2
        float dx = (x1b + x2b - x1a - x2a) * 0.5f;
        float dy = (y1b + y2b - y1a - y2a) * 0.5f;
        float rho2 = dx * dx + dy * dy;

        // Aspect-ratio consistency
        float dat = atanf(wb * __builtin_amdgcn_rcpf(hb + EPSF))
                  - atanf(wa * __builtin_amdgcn_rcpf(ha + EPSF));
        float v = INV_PI2 * dat * dat;
        float alpha = v * __builtin_amdgcn_rcpf(1.0f - iou + v + EPSF);

        float ciou = iou - rho2 * __builtin_amdgcn_rcpf(c2) - alpha * v;

        bool valid = (msk[i] != 0);
        float iou_m = valid ? ciou : 0.0f;
        __builtin_nontemporal_store(iou_m, &iou_out[i]);

        // (1 - iou_masked) * box_norm * m  ==  valid ? (1 - ciou) * bnorm : 0
        acc += valid ? (1.0f - ciou) * bnorm[i] : 0.0f;
    }

    // Block reduction: 8 wave32 waves -> LDS -> wave0
    __shared__ float lds[8];
    float w = wave_reduce_add(acc);
    const int lane = threadIdx.x & 31;
    const int wid  = threadIdx.x >> 5;
    if (lane == 0) lds[wid] = w;
    __syncthreads();
    if (wid == 0) {
        float s = (lane < 8) ? lds[lane] : 0.0f;
        s = wave_reduce_add(s);
        if (lane == 0) partials[blockIdx.x] = s;
    }
}

__global__ __launch_bounds__(256)
void ciou_finalize_kernel(const float* __restrict__ partials, int np,
                          const float* __restrict__ cls_norm,
                          float*       __restrict__ out_loss)
{
    float s = 0.0f;
    for (int i = threadIdx.x; i < np; i += 256)
        s += partials[i];

    __shared__ float lds[8];
    float w = wave_reduce_add(s);
    const int lane = threadIdx.x & 31;
    const int wid  = threadIdx.x >> 5;
    if (lane == 0) lds[wid] = w;
    __syncthreads();
    if (threadIdx.x == 0) {
        float t = 0.0f;
#pragma unroll
        for (int k = 0; k < 8; ++k) t += lds[k];
        out_loss[0] = t / cls_norm[0];
    }
}

extern "C" void kernel_launch(void* const* d_in, const int* in_sizes, int n_in,
                              void* d_out, int out_size, void* d_ws, size_t ws_size,
                              hipStream_t stream)
{
    // setup_inputs order: predicts_bbox, targets_bbox, valid_masks, box_norm, cls_norm
    const f32x4* pb    = (const f32x4*)d_in[0];
    const f32x4* tb    = (const f32x4*)d_in[1];
    const int*   msk   = (const int*)  d_in[2];
    const float* bnorm = (const float*)d_in[3];
    const float* cls   = (const float*)d_in[4];

    float* out = (float*)d_out;            // out[0] = loss, out[1..] = iou
    const int n = in_sizes[2];             // B * A elements

    int nblk = 2048;                       // 2048 blocks x 8 waves = plenty of latency hiding
    int maxp = (int)(ws_size / sizeof(float));
    if (maxp < nblk) nblk = (maxp > 0) ? maxp : 1;
    float* partials = (float*)d_ws;

    ciou_main_kernel<<<nblk, 256, 0, stream>>>(pb, tb, msk, bnorm,
                                               out + 1, partials, n);
    ciou_finalize_kernel<<<1, 256, 0, stream>>>(partials, nblk, cls, out);
}